// LinearSVI_30631706755327
// MI455X (gfx1250) — compile-verified
//
#include <hip/hip_runtime.h>
#include <hip/hip_bf16.h>

// LinearSVI on MI455X (gfx1250).
// Memory-bound streaming GEMV: [s q] per row via V_WMMA_F32_16X16X4_F32,
// exact f32 accumulation. 512 MB single pass -> ~22 us floor at 23.3 TB/s.

typedef __attribute__((ext_vector_type(2))) float v2f;
typedef __attribute__((ext_vector_type(8))) float v8f;

#define LATENT 512
#define TILE_ROWS 16
#define WAVES_PER_BLOCK 8

__global__ __launch_bounds__(256) void linear_svi_wmma(
    const float* __restrict__ data, const float* __restrict__ z,
    const float* __restrict__ muP, const float* __restrict__ tauP,
    float* __restrict__ outMu, float* __restrict__ outTau, int rows)
{
    const int lane = threadIdx.x & 31;
    const int wave = threadIdx.x >> 5;
    int rbase = (blockIdx.x * WAVES_PER_BLOCK + wave) * TILE_ROWS;
    if (rbase > rows - TILE_ROWS) rbase = rows - TILE_ROWS; // tail clamp (dup work, same values)

    const int m     = lane & 15;       // A: row-in-tile   | B: column N
    const int khalf = (lane >> 4) * 2; // A/B K sub-offset: lanes 0-15 -> K{0,1}, 16-31 -> K{2,3}

    const float* p  = data + (size_t)(rbase + m) * LATENT + khalf;
    const float* zp = z + khalf;

    v8f accS = {};  // column 0 accumulates s = data . z
    v8f accQ = {};  // column 0 accumulates q = (data^2) . (z^2)

    for (int k0 = 0; k0 < LATENT; k0 += 32) {
        __builtin_prefetch(p + k0 + 256, 0, 0); // ~1KB ahead in the row stream
#pragma unroll
        for (int kk = 0; kk < 32; kk += 4) {
            v2f a  = *(const v2f*)(p  + k0 + kk);  // global_load_b64, 8B aligned
            v2f zv = *(const v2f*)(zp + k0 + kk);  // hot in WGP$/L2 (2KB total)

            v2f a2; a2.x = a.x * a.x; a2.y = a.y * a.y;

            // B fragments: only column N==0 nonzero
            v2f bs, bq;
            bs.x = (m == 0) ? zv.x : 0.0f;
            bs.y = (m == 0) ? zv.y : 0.0f;
            bq.x = (m == 0) ? zv.x * zv.x : 0.0f;
            bq.y = (m == 0) ? zv.y * zv.y : 0.0f;

            // D = A x B + C (f32, exact). Two independent accumulator chains.
            accS = __builtin_amdgcn_wmma_f32_16x16x4_f32(
                false, a,  false, bs, (short)0, accS, false, false);
            accQ = __builtin_amdgcn_wmma_f32_16x16x4_f32(
                false, a2, false, bq, (short)0, accQ, false, false);
        }
    }

    // D layout: VGPR g: lanes 0-15 = D[M=g][N=lane], lanes 16-31 = D[M=8+g][N=lane-16]
    // => s,q for row g at lane 0; row 8+g at lane 16. Gather so lanes 0..15 own one row each.
    float s = 0.0f, q = 0.0f;
#pragma unroll
    for (int g = 0; g < 8; ++g) {
        float sl = __shfl(accS[g], 0, 32);
        float ql = __shfl(accQ[g], 0, 32);
        float sh = __shfl(accS[g], 16, 32);
        float qh = __shfl(accQ[g], 16, 32);
        if (lane == g)     { s = sl; q = ql; }
        if (lane == g + 8) { s = sh; q = qh; }
    }

    if (lane < 16) {
        const float mu  = muP[0];
        const float tau = tauP[0];
        const int row = rbase + lane;
        float irMu = (s + mu) * (1.0f / (1.0f + (float)LATENT));
        float dm   = irMu - mu;
        float sq   = q - 2.0f * mu * s + (float)LATENT * mu * mu; // sum((xz-mu)^2)
        float irTau = (tau + 0.5f * (float)LATENT) / (1.0f + 0.5f * sq + dm * dm);
        outMu[row]  = irMu;
        outTau[row] = irTau;
    }
}

extern "C" void kernel_launch(void* const* d_in, const int* in_sizes, int n_in,
                              void* d_out, int out_size, void* d_ws, size_t ws_size,
                              hipStream_t stream) {
    const float* data = (const float*)d_in[0];
    const float* z    = (const float*)d_in[1];
    const float* mu   = (const float*)d_in[2];
    const float* tau  = (const float*)d_in[3];

    const int rows = in_sizes[0] / LATENT;      // 262144
    float* outMu  = (float*)d_out;              // first `rows` floats
    float* outTau = outMu + rows;               // next `rows` floats

    const int rowsPerBlock = TILE_ROWS * WAVES_PER_BLOCK; // 128
    const int blocks = (rows + rowsPerBlock - 1) / rowsPerBlock; // 2048

    linear_svi_wmma<<<blocks, 256, 0, stream>>>(data, z, mu, tau, outMu, outTau, rows);
}